// CTCContrastLoss_67869073211889
// MI455X (gfx1250) — compile-verified
//
#include <hip/hip_runtime.h>
#include <hip/hip_bf16.h>
#include <stdint.h>

#define NEGF  (-1e30f)
#define EPSF  (1e-6f)
#define MARGINF (16.0f)
#define CMAX  256   // C in reference; lp row buffer size
#define LMAX  208   // >= 2 + L (L = 2*S+1 = 201)
#define NBUF  4     // emission-row ring depth (prefetch distance 3)

// ---- CDNA5 async global->LDS data mover ----
// Hard-require on the DEVICE pass only; the host pass (x86) doesn't have
// AMDGCN target builtins and must not trip the check.
#if defined(__HIP_DEVICE_COMPILE__)
#  if !defined(__has_builtin) || \
      !__has_builtin(__builtin_amdgcn_global_load_async_to_lds_b128) || \
      !__has_builtin(__builtin_amdgcn_s_wait_asynccnt)
#    error "gfx1250 async global->LDS builtins not available on device pass"
#  endif
#  define WAIT_ASYNC(n) __builtin_amdgcn_s_wait_asynccnt(n)
#else
#  define WAIT_ASYNC(n) ((void)0)
#endif

// Exact parameter types per hipcc diagnostic (round 1):
//   arg0: v4i addrspace(1)*, arg1: v4i addrspace(3)*, then imm offset, imm cpol.
typedef int v4i __attribute__((vector_size(16)));
typedef __attribute__((address_space(1))) v4i g_v4i;
typedef __attribute__((address_space(3))) v4i l_v4i;

__device__ __forceinline__ void async_copy_16B(const float* g, float* l) {
#if defined(__HIP_DEVICE_COMPILE__)
  __builtin_amdgcn_global_load_async_to_lds_b128((g_v4i*)(g), (l_v4i*)(l), 0, 0);
#else
  float4 v = *(const float4*)g;   // host-pass stub; never executed
  *(float4*)l = v;
#endif
}

__device__ __forceinline__ float logadd3(float a, float b, float c) {
  float m = fmaxf(a, fmaxf(b, c));
  float s = __expf(a - m) + __expf(b - m) + __expf(c - m);
  return m + __logf(s);
}

// One block per sample n; threads [0,256) -> label set 0, [256,512) -> ref set.
// Thread handles extended position s = tid & 255 (active if s < L).
__global__ __launch_bounds__(512)
void ctc_alpha_kernel(const float* __restrict__ lp,       // (T,N,C)
                      const int* __restrict__ labels,     // (N,S)
                      const int* __restrict__ ref_labels, // (N,S)
                      const int* __restrict__ in_len,     // (N)
                      const int* __restrict__ lab_len,    // (N)
                      const int* __restrict__ ref_len,    // (N)
                      float* __restrict__ out_nll,        // (2,N) in workspace
                      int T, int N, int C, int S) {
  const int n   = blockIdx.x;
  const int tid = threadIdx.x;
  const int L   = 2 * S + 1;
  const int set = tid >> 8;
  const int s   = tid & 255;

  __shared__ float lp_buf[NBUF][CMAX];   // ring of emission rows
  __shared__ float alphab[2][2][LMAX];   // [ping][set][2 (-inf guard) + L]

  // Per-thread constants: extended symbol + skip-transition flag.
  const int* lab = (set == 0) ? labels : ref_labels;
  int  c_sym = 0;
  bool skip  = false;
  if (s < L && (s & 1)) {
    c_sym = lab[n * S + (s >> 1)];
    if (s >= 3) {
      int prv = lab[n * S + (s >> 1) - 1];
      skip = (c_sym != prv) && (c_sym != 0);
    }
  }
  const int len_in = in_len[n];
  const int tid4   = tid * 4;
  const bool loader = (tid4 < C);               // waves 0..1 issue the DMA
  const size_t rowStride = (size_t)N * C;
  const float* g0 = lp + (size_t)n * C;         // row t at g0 + t*rowStride

  // Prologue: issue rows 0..2, wait for row 0, publish, init alpha(t=0).
  if (loader) {
    async_copy_16B(g0 + tid4, &lp_buf[0][tid4]);
    if (T > 1) async_copy_16B(g0 + rowStride + tid4, &lp_buf[1][tid4]);
    if (T > 2) async_copy_16B(g0 + 2 * rowStride + tid4, &lp_buf[2][tid4]);
    if (T > 2)      WAIT_ASYNC(2);   // in-order completion => row 0 landed
    else if (T > 1) WAIT_ASYNC(1);
    else            WAIT_ASYNC(0);
  }
  __syncthreads();
  if (s < 2) { alphab[0][set][s] = NEGF; alphab[1][set][s] = NEGF; }  // guards
  if (s < L) {
    float v = NEGF;
    if (s == 0) v = lp_buf[0][0];
    else if (s == 1) v = lp_buf[0][c_sym];
    alphab[0][set][2 + s] = v;
  }

  float* ar = &alphab[0][set][0];   // alpha(t-1)
  float* aw = &alphab[1][set][0];   // alpha(t) destination

  for (int t = 1; t < T; ++t) {
    if (loader) {
      // Outstanding rows are a subset of {t, t+1, t+2}; async loads complete
      // in order, so <= (#rows younger than t) outstanding => row t landed.
      int rem = T - 1 - t;
      if (rem >= 2)      WAIT_ASYNC(2);
      else if (rem == 1) WAIT_ASYNC(1);
      else               WAIT_ASYNC(0);
    }
    __syncthreads();   // publishes row t + alpha(t-1); all compute(t-1) done
    if (loader) {
      int tf = t + 3;  // refill the slot last read at compute(t-1)
      if (tf < T)
        async_copy_16B(g0 + (size_t)tf * rowStride + tid4,
                       &lp_buf[tf & (NBUF - 1)][tid4]);
    }
    if (s < L) {
      float a0 = ar[2 + s];
      float a1 = ar[1 + s];
      float a2 = skip ? ar[s] : NEGF;
      float emit = lp_buf[t & (NBUF - 1)][c_sym];
      float nv = logadd3(a0, a1, a2) + emit;
      aw[2 + s] = (t < len_in) ? nv : a0;   // freeze past input_length
    }
    float* tmp = ar; ar = aw; aw = tmp;
  }
  __syncthreads();

  if (s == 0) {  // one thread per (sample, set); ar == last written buffer
    int ll  = ((set == 0) ? lab_len : ref_len)[n];
    int idx = 2 * ll;
    float hi = ar[2 + idx];
    float lo = ar[1 + idx];
    float m  = fmaxf(hi, lo);
    float v  = m + __logf(__expf(hi - m) + __expf(lo - m));
    out_nll[set * N + n] = -v;
  }
}

__global__ void ctc_reduce_kernel(const float* __restrict__ ctc,
                                  const float* __restrict__ ref,
                                  float* __restrict__ out, int N) {
  __shared__ float s_sum[256];
  __shared__ float s_hng[256];
  int i = threadIdx.x;
  float acc = 0.0f, hng = 0.0f;
  for (int k = i; k < N; k += blockDim.x) {
    float cv = ctc[k], rv = ref[k];
    acc += cv;
    float d_ap = fabsf(EPSF - cv);
    float d_an = fabsf(EPSF - rv);
    hng += fmaxf(d_ap - d_an + MARGINF, 0.0f);
  }
  s_sum[i] = acc; s_hng[i] = hng;
  __syncthreads();
  for (int off = blockDim.x >> 1; off > 0; off >>= 1) {
    if (i < off) { s_sum[i] += s_sum[i + off]; s_hng[i] += s_hng[i + off]; }
    __syncthreads();
  }
  if (i == 0) out[0] = s_sum[0] + s_hng[0] / (float)N;
}

extern "C" void kernel_launch(void* const* d_in, const int* in_sizes, int n_in,
                              void* d_out, int out_size, void* d_ws, size_t ws_size,
                              hipStream_t stream) {
  const float* predicts  = (const float*)d_in[0];  // (T,N,C) fp32 log-probs
  const int* labels      = (const int*)d_in[1];    // (N,S)
  const int* ref_labels  = (const int*)d_in[2];    // (N,S)
  const int* preds_len   = (const int*)d_in[3];    // (N)
  const int* lab_len     = (const int*)d_in[4];    // (N)
  const int* ref_len     = (const int*)d_in[5];    // (N)

  const int N = in_sizes[3];
  const int S = in_sizes[1] / N;
  const int C = CMAX;                                   // 256 classes (reference)
  const int T = (int)(in_sizes[0] / ((long long)N * C));

  float* ws_ctc = (float*)d_ws;        // (N)
  float* ws_ref = ws_ctc + N;          // (N)

  ctc_alpha_kernel<<<N, 512, 0, stream>>>(predicts, labels, ref_labels,
                                          preds_len, lab_len, ref_len,
                                          ws_ctc, T, N, C, S);
  ctc_reduce_kernel<<<1, 256, 0, stream>>>(ws_ctc, ws_ref, (float*)d_out, N);
}